// simple3Dloss_15040975470951
// MI455X (gfx1250) — compile-verified
//
#include <hip/hip_runtime.h>
#include <stdint.h>

#define TSTEPS 1000
#define WCOLS  256
#define RPB    4              // rows (waves) per block
#define NT     (RPB * 32)     // 128 threads, wave32

#define GLOBAL_AS __attribute__((address_space(1)))
#define LOCAL_AS  __attribute__((address_space(3)))

typedef int v4i __attribute__((vector_size(4 * sizeof(int))));

// ---------------------------------------------------------------------------
// Async global -> LDS staging of 2 x 16B per lane (one 1KB row per wave via
// the instruction's shared immediate offset applied to both addresses).
// ---------------------------------------------------------------------------
__device__ __forceinline__ void stage_row_async(const float* g, float* l, int lane) {
#if defined(__AMDGCN__)
#if __has_builtin(__builtin_amdgcn_global_load_async_to_lds_b128)
  __builtin_amdgcn_global_load_async_to_lds_b128(
      (GLOBAL_AS v4i*)(g + lane * 4), (LOCAL_AS v4i*)(l + lane * 4), 0, 0);
  __builtin_amdgcn_global_load_async_to_lds_b128(
      (GLOBAL_AS v4i*)(g + lane * 4), (LOCAL_AS v4i*)(l + lane * 4), 512, 0);
#else
  unsigned      l0 = (unsigned)(uintptr_t)(LOCAL_AS void*)(l + lane * 4);
  unsigned long long g0 = (unsigned long long)(uintptr_t)(g + lane * 4);
  asm volatile("global_load_async_to_lds_b128 %0, %1, off"
               :: "v"(l0), "v"(g0) : "memory");
  asm volatile("global_load_async_to_lds_b128 %0, %1, off offset:512"
               :: "v"(l0), "v"(g0) : "memory");
#endif
#else
  // host-compile placeholder (never executed on device)
  for (int i = 0; i < 4; ++i) {
    l[lane * 4 + i]       = g[lane * 4 + i];
    l[lane * 4 + 128 + i] = g[lane * 4 + 128 + i];
  }
#endif
}

__device__ __forceinline__ void wait_async_zero() {
#if defined(__AMDGCN__)
#if __has_builtin(__builtin_amdgcn_s_wait_asynccnt)
  __builtin_amdgcn_s_wait_asynccnt(0);
#else
  asm volatile("s_wait_asynccnt 0" ::: "memory");
#endif
#endif
}

// ---------------------------------------------------------------------------
// One wave per (b,h) row: stage 2x256 floats to LDS asynchronously while
// zeroing 2x1000 u32 bins (b128 stores), scatter-max column indices with
// ds_max_u32, then accumulate sum((rec_bins - tgt_bins)^2) exactly in double
// reading bins back as b128.
// ---------------------------------------------------------------------------
__global__ __launch_bounds__(NT) void holo_mse_kernel(
    const float* __restrict__ rec, const float* __restrict__ tgt,
    double* __restrict__ partials, int rows)
{
  __shared__ __align__(16) unsigned bins[RPB][2][TSTEPS];   // 32000 B
  __shared__ __align__(16) float    stage[RPB][2][WCOLS];   //  8192 B
  __shared__ double wpart[RPB];

  const int tid  = threadIdx.x;
  const int lane = tid & 31;
  const int wv   = tid >> 5;
  const int row  = blockIdx.x * RPB + wv;
  const bool valid = (row < rows);

  // 1) kick off async staging of this wave's two rows
  if (valid) {
    stage_row_async(rec + (size_t)row * WCOLS, &stage[wv][0][0], lane);
    stage_row_async(tgt + (size_t)row * WCOLS, &stage[wv][1][0], lane);
  }

  // 2) zero all bins cooperatively (vectorized) while the DMA is in flight
  {
    uint4* bp4 = (uint4*)&bins[0][0][0];            // 2000 x uint4
    const uint4 z4 = make_uint4(0u, 0u, 0u, 0u);
    for (int i = tid; i < (RPB * 2 * TSTEPS) / 4; i += NT) bp4[i] = z4;
  }
  wait_async_zero();
  __syncthreads();

  // 3) quantize + scatter-max column index (matches JAX semantics exactly)
  if (valid) {
#pragma unroll
    for (int k = 0; k < WCOLS / 32; ++k) {
      const int c = k * 32 + lane;
#pragma unroll
      for (int img = 0; img < 2; ++img) {
        const float x = stage[wv][img][c];
        if (x != 0.0f) {                       // x==0 -> index TSTEPS -> dropped
          int q = (int)(x * 1000.0f) - 1;      // f32 mul, truncate toward zero
          q %= TSTEPS;
          if (q < 0) q += TSTEPS;              // Python-style mod
          atomicMax(&bins[wv][img][q], (unsigned)c);   // ds_max_u32
        }
      }
    }
  }
  __syncthreads();

  // 4) exact squared-diff accumulation (all terms are small integers)
  double s = 0.0;
  if (valid) {
    const uint4* a4 = (const uint4*)&bins[wv][0][0];   // 250 x uint4
    const uint4* b4 = (const uint4*)&bins[wv][1][0];
    for (int i = lane; i < TSTEPS / 4; i += 32) {
      const uint4 a = a4[i];
      const uint4 b = b4[i];
      const float d0 = (float)a.x - (float)b.x;
      const float d1 = (float)a.y - (float)b.y;
      const float d2 = (float)a.z - (float)b.z;
      const float d3 = (float)a.w - (float)b.w;
      s += (double)(d0 * d0) + (double)(d1 * d1) +
           (double)(d2 * d2) + (double)(d3 * d3);
    }
  }
#pragma unroll
  for (int o = 16; o > 0; o >>= 1) s += __shfl_down(s, o, 32);
  if (lane == 0) wpart[wv] = s;
  __syncthreads();
  if (tid == 0) {
    double t = 0.0;
#pragma unroll
    for (int i = 0; i < RPB; ++i) t += wpart[i];
    partials[blockIdx.x] = t;                  // deterministic: no atomics
  }
}

// ---------------------------------------------------------------------------
// Reduce block partials and write the f32 mean.
// ---------------------------------------------------------------------------
__global__ __launch_bounds__(256) void finalize_kernel(
    const double* __restrict__ partials, int nparts,
    float* __restrict__ out, double inv_n)
{
  __shared__ double sh[256];
  double s = 0.0;
  for (int i = threadIdx.x; i < nparts; i += 256) s += partials[i];
  sh[threadIdx.x] = s;
  __syncthreads();
  for (int o = 128; o > 0; o >>= 1) {
    if ((int)threadIdx.x < o) sh[threadIdx.x] += sh[threadIdx.x + o];
    __syncthreads();
  }
  if (threadIdx.x == 0) out[0] = (float)(sh[0] * inv_n);
}

extern "C" void kernel_launch(void* const* d_in, const int* in_sizes, int n_in,
                              void* d_out, int out_size, void* d_ws, size_t ws_size,
                              hipStream_t stream) {
  const float* rec = (const float*)d_in[0];
  const float* tgt = (const float*)d_in[1];
  float* out = (float*)d_out;

  const int n      = in_sizes[0];        // 64*1*256*256 = 4194304
  const int rows   = n / WCOLS;          // 16384
  const int blocks = (rows + RPB - 1) / RPB;   // 4096
  double* partials = (double*)d_ws;      // 4096 doubles = 32 KB scratch

  holo_mse_kernel<<<blocks, NT, 0, stream>>>(rec, tgt, partials, rows);

  const double inv_n = 1.0 / ((double)rows * (double)TSTEPS);
  finalize_kernel<<<1, 256, 0, stream>>>(partials, blocks, out, inv_n);
}